// rgcn_layer_67525475828084
// MI455X (gfx1250) — compile-verified
//
#include <hip/hip_runtime.h>

#define NNODES 100000
#define NREL   4
#define NEDGES 500000
#define D      128

typedef __attribute__((ext_vector_type(2))) float v2f;
typedef __attribute__((ext_vector_type(8))) float v8f;

// One wave (32 lanes) per edge: lane gathers float4 of the src row,
// atomically adds into the dst row. Lane 0 counts the degree.
__global__ __launch_bounds__(256) void rgcn_scatter(
    const float* __restrict__ x,
    const int*   __restrict__ src,
    const int*   __restrict__ dst,
    float*       __restrict__ agg,
    float*       __restrict__ deg,
    int nEdges)
{
    int lane = threadIdx.x & 31;
    int e = blockIdx.x * 8 + (threadIdx.x >> 5);
    if (e >= nEdges) return;

    int s = src[e];
    int d = dst[e];

    const float4* xr = (const float4*)(x + (size_t)s * D);
    float4 v = xr[lane];

    float* ar = agg + (size_t)d * D + lane * 4;
    atomicAdd(ar + 0, v.x);
    atomicAdd(ar + 1, v.y);
    atomicAdd(ar + 2, v.z);
    atomicAdd(ar + 3, v.w);

    if (lane == 0) atomicAdd(deg + d, 1.0f);
}

// out[tile] (+)= (agg/deg) @ W + b, using V_WMMA_F32_16X16X4_F32.
// Block = 16 rows x 128 cols; 8 waves, each computing one 16x16 tile.
// first: initialize accumulator from bias only. last: apply ReLU on store.
__global__ __launch_bounds__(256) void rgcn_gemm(
    const float* __restrict__ agg,
    const float* __restrict__ deg,
    const float* __restrict__ Wr,
    const float* __restrict__ bias,
    float*       __restrict__ out,
    int first, int last)
{
    int lane = threadIdx.x & 31;
    int wv   = threadIdx.x >> 5;      // 0..7 -> column tile
    int m    = lane & 15;             // row (A) / column (B,C,D) within tile
    int kh   = lane >> 4;             // K half-select (0/1)

    int row0 = blockIdx.x * 16;
    int n0   = wv * 16;
    int col  = n0 + m;

    // norm='right': scale A rows by 1/max(deg,1). Constant per (lane row).
    float scale = 1.0f / fmaxf(deg[row0 + m], 1.0f);
    float bv    = bias[col];

    // C/D layout: VGPR i, lanes 0-15 -> M=i; lanes 16-31 -> M=i+8; N = lane&15.
    v8f c;
    if (first) {
#pragma unroll
        for (int i = 0; i < 8; ++i) c[i] = bv;
    } else {
#pragma unroll
        for (int i = 0; i < 8; ++i)
            c[i] = out[(size_t)(row0 + i + 8 * kh) * D + col] + bv;
    }

    // A layout (16x4 f32): lane L holds A[L&15, 2*(L>>4)+j], j=0,1 -> float2 load.
    const float* arow = agg + (size_t)(row0 + m) * D + 2 * kh;
    // B layout (4x16 f32): lane L holds B[2*(L>>4)+j, L&15].
    const float* wb = Wr + (size_t)(2 * kh) * D + col;

#pragma unroll 4
    for (int k0 = 0; k0 < D; k0 += 4) {
        float2 av = *(const float2*)(arow + k0);
        v2f a;
        a[0] = av.x * scale;
        a[1] = av.y * scale;

        v2f bm;
        bm[0] = wb[(size_t)(k0 + 0) * D];
        bm[1] = wb[(size_t)(k0 + 1) * D];

        // (neg_a, A, neg_b, B, c_mod, C, reuse_a, reuse_b)
        c = __builtin_amdgcn_wmma_f32_16x16x4_f32(
                false, a, false, bm, (short)0, c, false, false);
    }

#pragma unroll
    for (int i = 0; i < 8; ++i) {
        float v = c[i];
        if (last) v = fmaxf(v, 0.0f);
        out[(size_t)(row0 + i + 8 * kh) * D + col] = v;
    }
}

extern "C" void kernel_launch(void* const* d_in, const int* in_sizes, int n_in,
                              void* d_out, int out_size, void* d_ws, size_t ws_size,
                              hipStream_t stream)
{
    const float* x   = (const float*)d_in[0];   // [N, 128]
    const float* W   = (const float*)d_in[1];   // [4, 128, 128]
    const float* b   = (const float*)d_in[2];   // [4, 128]
    const int*   src = (const int*)  d_in[3];   // [4, E]
    const int*   dst = (const int*)  d_in[4];   // [4, E]
    float*       out = (float*)d_out;           // [N, 128]

    float* agg = (float*)d_ws;                     // N*128 floats
    float* deg = agg + (size_t)NNODES * D;         // N floats
    size_t zeroBytes = ((size_t)NNODES * D + NNODES) * sizeof(float);

    for (int r = 0; r < NREL; ++r) {
        hipMemsetAsync(agg, 0, zeroBytes, stream);
        rgcn_scatter<<<(NEDGES + 7) / 8, 256, 0, stream>>>(
            x, src + (size_t)r * NEDGES, dst + (size_t)r * NEDGES,
            agg, deg, NEDGES);
        rgcn_gemm<<<NNODES / 16, 256, 0, stream>>>(
            agg, deg, W + (size_t)r * D * D, b + (size_t)r * D,
            out, r == 0 ? 1 : 0, r == NREL - 1 ? 1 : 0);
    }
}